// RN_9620726743791
// MI455X (gfx1250) — compile-verified
//
#include <hip/hip_runtime.h>

typedef __attribute__((ext_vector_type(2))) float v2f;
typedef __attribute__((ext_vector_type(8))) float v8f;

static constexpr int DIM = 256;   // D
static constexpr int SEQ = 256;   // S
static constexpr int NB  = 16;    // B
static constexpr int ROWS = NB * SEQ; // 4096 gathered rows

// ---------------------------------------------------------------------------
// Kernel 1: L[m,e] = sum_d emb[X[m],d] * Wl[e,d] + bl[e];  same for R.
// One wave per 16x16 tile; both L and R tiles computed so the gathered
// A-tile (emb rows) is loaded once and fed to two WMMA chains.
// f32 WMMA layout (ISA 7.12.2):
//   A 16x4 : lane c (0-15) row M=c, VGPR0/1 = K0/K1; lanes 16-31 = K2/K3
//   B 4x16 : VGPR0 lanes0-15 = K0 row, lanes16-31 = K2; VGPR1 = K1/K3
//   C 16x16: VGPR r, lanes0-15 -> (M=r, N=lane); lanes16-31 -> (M=r+8)
// ---------------------------------------------------------------------------
__global__ __launch_bounds__(256) void k_project(
    const int* __restrict__ X, const float* __restrict__ emb,
    const float* __restrict__ Wl, const float* __restrict__ bl,
    const float* __restrict__ Wr, const float* __restrict__ br,
    float* __restrict__ L, float* __restrict__ R)
{
  const int wave = threadIdx.x >> 5;
  const int lane = threadIdx.x & 31;
  const int gid  = blockIdx.x * 8 + wave;       // 4096 tiles total
  const int m0   = (gid >> 4) << 4;             // 256 row tiles
  const int n0   = (gid & 15) << 4;             // 16 col tiles
  const int c    = lane & 15;
  const int koff = (lane >> 4) * 2;             // K pair select per half-wave

  const int tok = X[m0 + c];
  const float* __restrict__ arow  = emb + (size_t)tok * DIM;
  const float* __restrict__ wlrow = Wl + (size_t)(n0 + c) * DIM;
  const float* __restrict__ wrrow = Wr + (size_t)(n0 + c) * DIM;

  v8f accL = {};
  v8f accR = {};
  #pragma unroll 4
  for (int k0 = 0; k0 < DIM; k0 += 4) {
    v2f a, bbl, bbr;
    a.x   = arow[k0 + koff];
    a.y   = arow[k0 + koff + 1];
    bbl.x = wlrow[k0 + koff];
    bbl.y = wlrow[k0 + koff + 1];
    bbr.x = wrrow[k0 + koff];
    bbr.y = wrrow[k0 + koff + 1];
    accL = __builtin_amdgcn_wmma_f32_16x16x4_f32(false, a, false, bbl,
                                                 (short)0, accL, false, false);
    accR = __builtin_amdgcn_wmma_f32_16x16x4_f32(false, a, false, bbr,
                                                 (short)0, accR, false, false);
  }

  const float biasL = bl[n0 + c];
  const float biasR = br[n0 + c];
  const int mrow = m0 + (lane >> 4) * 8;
  #pragma unroll
  for (int r = 0; r < 8; ++r) {
    L[(size_t)(mrow + r) * DIM + n0 + c] = accL[r] + biasL;
    R[(size_t)(mrow + r) * DIM + n0 + c] = accR[r] + biasR;
  }
}

// ---------------------------------------------------------------------------
// Kernel 2: partial pairwise relu sums via relu(x) = 0.5*(x + |x|).
//   sum_{i in chunk, j} relu(l_i + r_j)
//     = 0.5*( S*sum_i l_i + 32*sum_j r_j + sum_{i,j} |l_i + r_j| )
// Linear part hoisted out of the S^2 loop; |.| is a free VALU source
// modifier, so the hot loop is 2 VALU ops/element instead of 3.
// grid = (b in 0..15, i-chunk in 0..7), thread = d (coalesced over lanes).
// ---------------------------------------------------------------------------
__global__ __launch_bounds__(256) void k_pairsum(
    const float* __restrict__ L, const float* __restrict__ R,
    float* __restrict__ P)
{
  const int b  = blockIdx.x;
  const int ic = blockIdx.y;
  const int d  = threadIdx.x;

  const float* __restrict__ Lp = L + ((size_t)b * SEQ + ic * 32) * DIM + d;
  const float* __restrict__ Rp = R + (size_t)b * SEQ * DIM + d;

  float l[32];
  float sl = 0.0f;
  #pragma unroll
  for (int i = 0; i < 32; ++i) {
    l[i] = Lp[(size_t)i * DIM];
    sl += l[i];
  }

  float accAbs = 0.0f;
  float sr = 0.0f;
  for (int j = 0; j < SEQ; ++j) {
    const float r = Rp[(size_t)j * DIM];
    sr += r;
    #pragma unroll
    for (int i = 0; i < 32; ++i) {
      accAbs += __builtin_fabsf(l[i] + r);  // v_add + v_add(|src|)
    }
  }
  const float total = 0.5f * (accAbs + (float)SEQ * sl + 32.0f * sr);
  P[((size_t)ic * NB + b) * DIM + d] = total;
}

// ---------------------------------------------------------------------------
// Kernel 3: reduce partials into Hm (LDS), then out = Hm * Wrel^T + brel.
// Single block, 512 threads = 16 waves; wave w computes N-tile w (M=16 = B).
// ---------------------------------------------------------------------------
__global__ __launch_bounds__(512) void k_finish(
    const float* __restrict__ P, const float* __restrict__ Wrel,
    const float* __restrict__ brel, float* __restrict__ out)
{
  __shared__ float hm[NB * DIM];
  const int tid = threadIdx.x;

  for (int e = tid; e < NB * DIM; e += 512) {
    float s = 0.0f;
    #pragma unroll
    for (int cch = 0; cch < 8; ++cch) s += P[(size_t)cch * NB * DIM + e];
    hm[e] = s * (1.0f / ((float)SEQ * (float)SEQ));
  }
  __syncthreads();

  const int wave = tid >> 5;
  const int lane = tid & 31;
  const int n0   = wave << 4;
  const int c    = lane & 15;
  const int koff = (lane >> 4) * 2;
  const float* __restrict__ wrow = Wrel + (size_t)(n0 + c) * DIM;

  v8f acc = {};
  #pragma unroll 4
  for (int k0 = 0; k0 < DIM; k0 += 4) {
    v2f a, bb;
    a.x  = hm[c * DIM + k0 + koff];
    a.y  = hm[c * DIM + k0 + koff + 1];
    bb.x = wrow[k0 + koff];
    bb.y = wrow[k0 + koff + 1];
    acc = __builtin_amdgcn_wmma_f32_16x16x4_f32(false, a, false, bb,
                                                (short)0, acc, false, false);
  }

  const float bias = brel[n0 + c];
  const int mrow = (lane >> 4) * 8;
  #pragma unroll
  for (int r = 0; r < 8; ++r)
    out[(size_t)(mrow + r) * DIM + n0 + c] = acc[r] + bias;
}

// ---------------------------------------------------------------------------
extern "C" void kernel_launch(void* const* d_in, const int* in_sizes, int n_in,
                              void* d_out, int out_size, void* d_ws, size_t ws_size,
                              hipStream_t stream) {
  (void)in_sizes; (void)n_in; (void)out_size; (void)ws_size;
  const int*   X    = (const int*)d_in[0];
  const float* emb  = (const float*)d_in[1];
  const float* Wl   = (const float*)d_in[2];
  const float* bl   = (const float*)d_in[3];
  const float* Wr   = (const float*)d_in[4];
  const float* br   = (const float*)d_in[5];
  const float* Wrel = (const float*)d_in[6];
  const float* brel = (const float*)d_in[7];
  float* out = (float*)d_out;

  float* ws = (float*)d_ws;
  float* L  = ws;                              // 4096*256 floats (4 MB)
  float* R  = L + (size_t)ROWS * DIM;          // 4 MB
  float* P  = R + (size_t)ROWS * DIM;          // 8*16*256 floats (128 KB)

  k_project<<<dim3(512), dim3(256), 0, stream>>>(X, emb, Wl, bl, Wr, br, L, R);
  k_pairsum<<<dim3(NB, 8), dim3(256), 0, stream>>>(L, R, P);
  k_finish <<<dim3(1),   dim3(512), 0, stream>>>(P, Wrel, brel, out);
}